// CNN_60868276519451
// MI455X (gfx1250) — compile-verified
//
#include <hip/hip_runtime.h>
#include <hip/hip_bf16.h>
#include <cmath>

typedef __attribute__((ext_vector_type(2))) float v2f;
typedef __attribute__((ext_vector_type(8))) float v8f;

#define C_IN   103
#define NPTS   9
#define KDIM   927      // C_IN * NPTS
#define KPAD   928      // KDIM padded to a multiple of 4 (zero row at 927)
#define SSTR   929      // per-pixel LDS row stride (odd -> bank-friendly)
#define TILE   16       // pixels per tile
#define H_IMG  32
#define W_IMG  32
#define COUT   32
#define NOFF   18       // 2*NPTS offset channels
#define BN_EPS 1e-5f

// ---------------------------------------------------------------------------
// Kernel 0: pad main conv weights (COUT, KDIM) -> (COUT, KPAD) with a zero
// column at k = 927 so the WMMA K-loop needs no tail guards and can use
// aligned float2 (global_load_b64) fragment loads.
// ---------------------------------------------------------------------------
__global__ __launch_bounds__(256) void pad_weights(
    const float* __restrict__ w, float* __restrict__ wpad)
{
    const int i = blockIdx.x * 256 + threadIdx.x;
    if (i < COUT * KPAD) {
        const int row = i / KPAD, k = i % KPAD;
        wpad[i] = (k < KDIM) ? w[row * KDIM + k] : 0.0f;
    }
}

// ---------------------------------------------------------------------------
// Kernel 1: offset-predicting 3x3 conv (pad=1). One block per (image, out-ch).
// Branchless zero-padding: index clamped via &31 (always a legal address in
// the channel plane), invalid taps selected to zero via cndmask.
// ---------------------------------------------------------------------------
__global__ __launch_bounds__(256) void deform_offset_conv(
    const float* __restrict__ x, const float* __restrict__ w_off,
    const float* __restrict__ b_off, float* __restrict__ off)
{
    __shared__ float wsh[KDIM];
    const int b  = blockIdx.x / NOFF;
    const int oc = blockIdx.x % NOFF;
    for (int i = threadIdx.x; i < KDIM; i += 256)
        wsh[i] = w_off[oc * KDIM + i];
    __syncthreads();

    const float bias = b_off[oc];
    const float* xb = x + (size_t)b * C_IN * (H_IMG * W_IMG);

    for (int pix = threadIdx.x; pix < H_IMG * W_IMG; pix += 256) {
        const int h = pix >> 5, w = pix & 31;
        float acc = bias;
        for (int c = 0; c < C_IN; ++c) {
            const float* xc = xb + c * (H_IMG * W_IMG);
            const float* wk = wsh + c * 9;
#pragma unroll
            for (int ky = 0; ky < 3; ++ky) {
                const int yy = h + ky - 1;
#pragma unroll
                for (int kx = 0; kx < 3; ++kx) {
                    const int xx = w + kx - 1;
                    const bool valid = ((unsigned)yy < (unsigned)H_IMG) &
                                       ((unsigned)xx < (unsigned)W_IMG);
                    const float xv = xc[(yy & 31) * W_IMG + (xx & 31)];
                    acc = fmaf(valid ? xv : 0.0f, wk[ky * 3 + kx], acc);
                }
            }
        }
        off[((size_t)b * NOFF + oc) * (H_IMG * W_IMG) + pix] = acc;
    }
}

// ---------------------------------------------------------------------------
// Kernel 2: bilinear sampling -> LDS, then 32x16 output tile GEMM via
// V_WMMA_F32_16X16X4_F32 (K padded to 928 -> 232 guard-free K-steps split
// across 4 waves), ds_add_f32 cross-wave reduce, fused bias + BN + ReLU.
// One 128-thread block per (image, 16-pixel tile).
// ---------------------------------------------------------------------------
__global__ __launch_bounds__(128) void deform_wmma_gemm(
    const float* __restrict__ x, const float* __restrict__ off,
    const float* __restrict__ wpad,   // (COUT, KPAD) row-major, k=927 zeroed
    const float* __restrict__ bias,
    const float* __restrict__ gamma, const float* __restrict__ beta,
    const float* __restrict__ rmean, const float* __restrict__ rvar,
    float* __restrict__ out)
{
    __shared__ float S[TILE * SSTR];          // S[p][k] sampled matrix
    __shared__ int   y0s[TILE * NPTS], x0s[TILE * NPTS];
    __shared__ float wys[TILE * NPTS], wxs[TILE * NPTS];
    __shared__ float accD[COUT * TILE];       // 32x16 output tile accumulator

    const int bb   = blockIdx.x >> 6;   // image index (64 tiles per image)
    const int tile = blockIdx.x & 63;
    const int tid  = threadIdx.x;
    const int p0   = tile * TILE;       // first linear pixel of tile

    for (int i = tid; i < COUT * TILE; i += 128) accD[i] = 0.0f;

    // ---- Phase A: sample coordinates for 16 pixels x 9 points -------------
    const float* offb = off + (size_t)bb * NOFF * (H_IMG * W_IMG);
    for (int i = tid; i < TILE * NPTS; i += 128) {
        const int p = i / NPTS, n = i % NPTS;
        const int pix = p0 + p;
        const int h = pix >> 5, w = pix & 31;
        const float oy = offb[n * (H_IMG * W_IMG) + pix];
        const float ox = offb[(NPTS + n) * (H_IMG * W_IMG) + pix];
        float py = (float)(h + 1) + (float)(n / 3 - 1) + oy;  // padded coords
        float px = (float)(w + 1) + (float)(n % 3 - 1) + ox;
        py = fminf(fmaxf(py, 0.0f), 33.0f);   // Hp-1 = 33
        px = fminf(fmaxf(px, 0.0f), 33.0f);
        int y0 = (int)floorf(py); y0 = min(max(y0, 0), 32);   // Hp-2 = 32
        int x0 = (int)floorf(px); x0 = min(max(x0, 0), 32);
        y0s[i] = y0; x0s[i] = x0;
        wys[i] = py - (float)y0;
        wxs[i] = px - (float)x0;
    }
    __syncthreads();

    // ---- Phase B: bilinear gather of all (c, n, p) into S ------------------
    const float* xb = x + (size_t)bb * C_IN * (H_IMG * W_IMG);
    for (int i = tid; i < C_IN * TILE * NPTS; i += 128) {
        const int c = i / (TILE * NPTS), r = i % (TILE * NPTS);
        const int p = r / NPTS, n = r % NPTS;
        const int k = c * NPTS + n;
        const int y0 = y0s[r], x0 = x0s[r];
        const float wy = wys[r], wx = wxs[r];
        const float* xc = xb + c * (H_IMG * W_IMG);
        // branchless padded fetch: clamped address, select zero on OOB
        auto fetch = [&](int yq, int xq) -> float {
            const int yi = yq - 1, xi = xq - 1;     // unpadded coords
            const bool valid = ((unsigned)yi < (unsigned)H_IMG) &
                               ((unsigned)xi < (unsigned)W_IMG);
            const float v = xc[(yi & 31) * W_IMG + (xi & 31)];
            return valid ? v : 0.0f;
        };
        const float v00 = fetch(y0,     x0    );
        const float v01 = fetch(y0,     x0 + 1);
        const float v10 = fetch(y0 + 1, x0    );
        const float v11 = fetch(y0 + 1, x0 + 1);
        const float sv = v00 * (1.0f - wy) * (1.0f - wx)
                       + v01 * (1.0f - wy) * wx
                       + v10 * wy * (1.0f - wx)
                       + v11 * wy * wx;
        S[p * SSTR + k] = sv;
    }
    // zero K-pad entry (k = 927) for all pixels
    for (int i = tid; i < TILE; i += 128) S[i * SSTR + KDIM] = 0.0f;
    __syncthreads();

    // ---- Phase C: guard-free WMMA f32 16x16x4 GEMM, 58 K-steps per wave ----
    const int wv   = tid >> 5;
    const int lane = tid & 31;
    const int mrow = lane & 15;          // A-matrix row / B-matrix pixel col
    const int koff = (lane >> 4) * 2;    // lanes 0-15: K k0,k0+1; 16-31: +2
    const float* wp0 = wpad + mrow * KPAD;          // Cout rows 0..15
    const float* wp1 = wpad + (mrow + 16) * KPAD;   // Cout rows 16..31
    const float* sp  = S + mrow * SSTR;
    v8f c0 = {};
    v8f c1 = {};
    const int ks_beg = wv * 58;          // 232 K-steps total (KPAD/4)
    const int ks_end = ks_beg + 58;
#pragma unroll 2
    for (int ks = ks_beg; ks < ks_end; ++ks) {
        const int k0 = ks * 4 + koff;    // even -> 8B-aligned float2 loads
        const v2f a0 = *(const v2f*)(wp0 + k0);
        const v2f a1 = *(const v2f*)(wp1 + k0);
        v2f bf;
        bf.x = sp[k0];
        bf.y = sp[k0 + 1];
        c0 = __builtin_amdgcn_wmma_f32_16x16x4_f32(
                 false, a0, false, bf, (short)0, c0, false, false);
        c1 = __builtin_amdgcn_wmma_f32_16x16x4_f32(
                 false, a1, false, bf, (short)0, c1, false, false);
    }

    // ---- reduce the 4 wave partials via LDS float atomics ------------------
#pragma unroll
    for (int v = 0; v < 8; ++v) {
        const int row = v + ((lane >> 4) << 3);  // C/D layout: VGPR v, lane hi
        const int col = lane & 15;
        atomicAdd(&accD[row * TILE + col],        c0[v]);
        atomicAdd(&accD[(row + 16) * TILE + col], c1[v]);
    }
    __syncthreads();

    // ---- fused bias + BatchNorm(eval) + ReLU + store -----------------------
    for (int i = tid; i < COUT * TILE; i += 128) {
        const int cout = i >> 4, col = i & 15;
        const float inv = gamma[cout] * rsqrtf(rvar[cout] + BN_EPS);
        float val = (accD[i] + bias[cout]) * inv + (beta[cout] - rmean[cout] * inv);
        val = fmaxf(val, 0.0f);
        out[((size_t)bb * COUT + cout) * (H_IMG * W_IMG) + p0 + col] = val;
    }
}

// ---------------------------------------------------------------------------
extern "C" void kernel_launch(void* const* d_in, const int* in_sizes, int n_in,
                              void* d_out, int out_size, void* d_ws, size_t ws_size,
                              hipStream_t stream) {
    const float* x     = (const float*)d_in[0];
    const float* w_off = (const float*)d_in[1];
    const float* b_off = (const float*)d_in[2];
    const float* w     = (const float*)d_in[3];   // (32, 103, 3, 3) == (32, 927)
    const float* b     = (const float*)d_in[4];
    const float* gamma = (const float*)d_in[5];
    const float* beta  = (const float*)d_in[6];
    const float* rmean = (const float*)d_in[7];
    const float* rvar  = (const float*)d_in[8];

    // workspace layout: [offsets (32*18*1024 f32)] [wpad (32*928 f32)]
    float* off  = (float*)d_ws;
    float* wpad = off + (size_t)32 * NOFF * (H_IMG * W_IMG);

    pad_weights<<<dim3((COUT * KPAD + 255) / 256), 256, 0, stream>>>(w, wpad);
    deform_offset_conv<<<dim3(32 * NOFF), 256, 0, stream>>>(x, w_off, b_off, off);
    deform_wmma_gemm<<<dim3(32 * 64), 128, 0, stream>>>(
        x, off, wpad, b, gamma, beta, rmean, rvar, (float*)d_out);
}